// YOLOXPostprocess_91336774517419
// MI455X (gfx1250) — compile-verified
//
#include <hip/hip_runtime.h>
#include <hip/hip_bf16.h>

// YOLOX postprocess for MI455X (gfx1250, wave32, 320KB LDS/WGP).
// Pipeline: init -> decode(3 levels) -> per-image NMS (1 WG/image).
// NMS keeps live scores + offset boxes register-resident (9 anchors/thread);
// LDS holds the async-DMA-staged offset boxes (for best-box broadcast) and
// tiny reduction scratch. 2 barriers per greedy iteration.

#define N_ANCH      8400
#define NUM_CLS     80
#define MAX_DET     100
#define NMS_THR     0.65f
#define SCORE_THR   0.01f
#define NMS_THREADS 1024
#define CHUNK       9            // ceil(8400 / 1024)
#define NEG_INF     (-__builtin_inff())

__device__ __forceinline__ float sigmoidf(float x) { return 1.0f / (1.0f + expf(-x)); }

// ---- gfx1250 async global->LDS path (guarded; falls back to plain copy) ----
#if defined(__HIP_DEVICE_COMPILE__) && defined(__gfx1250__) && \
    __has_builtin(__builtin_amdgcn_global_load_async_to_lds_b128)
#define YOLOX_ASYNC_LDS 1
// Builtin expects pointers to 16-byte int vectors (per hipcc diagnostic).
typedef int yolox_v4i __attribute__((vector_size(16)));

__device__ __forceinline__ __attribute__((address_space(3))) yolox_v4i*
yolox_to_lds(void* p) {
    // generic LDS pointer: low 32 bits are the LDS offset
    return (__attribute__((address_space(3))) yolox_v4i*)(unsigned long long)p;
}
__device__ __forceinline__ __attribute__((address_space(1))) yolox_v4i*
yolox_to_global(const void* p) {
    return (__attribute__((address_space(1))) yolox_v4i*)(unsigned long long)p;
}
#endif

// -------------------------------- init --------------------------------------
__global__ void yolox_init_kernel(unsigned int* offmax, int B) {
    int i = threadIdx.x;
    if (i < B) offmax[i] = 0u;  // bits of +0.0f
}

// ----------------------------- decode + score -------------------------------
// One thread per (image, anchor-in-level). Memory-bound: 85 coalesced f32
// loads/thread (80 cls + 4 reg + 1 obj). ~91MB total -> ~4us at 23.3 TB/s.
__global__ void yolox_decode_kernel(const float* __restrict__ cls,
                                    const float* __restrict__ reg,
                                    const float* __restrict__ obj,
                                    int hw, int w, float stride, int abase,
                                    float4*       __restrict__ boxes,
                                    float*        __restrict__ scores,
                                    int*          __restrict__ labels,
                                    float*        __restrict__ live,
                                    unsigned int* __restrict__ offmax)
{
    const int  a      = blockIdx.x * blockDim.x + threadIdx.x;
    const int  b      = blockIdx.y;
    const bool active = (a < hw);
    const int  aa     = active ? a : 0;

    const float* clsb = cls + (size_t)b * NUM_CLS * hw + aa;
    const float* regb = reg + (size_t)b * 4 * hw + aa;

    __builtin_prefetch(regb, 0, 3);                       // global_prefetch_b8
    __builtin_prefetch(obj + (size_t)b * hw + aa, 0, 3);

    // sigmoid is monotone: class argmax/max on logits (strict '>' = first max,
    // matching jnp.argmax semantics)
    float best = NEG_INF;
    int   lab  = 0;
#pragma unroll 8
    for (int c = 0; c < NUM_CLS; ++c) {
        float v = clsb[(size_t)c * hw];
        if (v > best) { best = v; lab = c; }
    }

    const float rx = regb[0];
    const float ry = regb[(size_t)hw];
    const float rw = regb[(size_t)2 * hw];
    const float rh = regb[(size_t)3 * hw];
    const float o  = obj[(size_t)b * hw + aa];

    const float sc = sigmoidf(best) * sigmoidf(o);

    const float gx  = (float)(aa % w) * stride;
    const float gy  = (float)(aa / w) * stride;
    const float xc  = rx * stride + gx;
    const float yc  = ry * stride + gy;
    const float wh2 = expf(rw) * stride * 0.5f;
    const float hh2 = expf(rh) * stride * 0.5f;
    const float x0 = xc - wh2, y0 = yc - hh2, x1 = xc + wh2, y1 = yc + hh2;

    float m = 0.0f;
    if (active) {
        const int gi = b * N_ANCH + abase + aa;
        boxes[gi]  = make_float4(x0, y0, x1, y1);
        scores[gi] = sc;
        labels[gi] = lab;
        live[gi]   = (sc >= SCORE_THR) ? sc : NEG_INF;
        m = fmaxf(fmaxf(fabsf(x0), fabsf(y0)), fmaxf(fabsf(x1), fabsf(y1)));
    }
    // per-image max|boxes| for the class-offset trick: wave32 shuffle reduce,
    // then one atomicMax/wave. Non-negative f32 compares correctly as u32 bits.
#pragma unroll
    for (int off = 16; off; off >>= 1) m = fmaxf(m, __shfl_down(m, off));
    if ((threadIdx.x & 31) == 0) atomicMax(offmax + b, __float_as_uint(m));
}

// ---------------------------------- NMS -------------------------------------
__global__ __launch_bounds__(NMS_THREADS)
void yolox_nms_kernel(const float4* __restrict__ boxes,
                      const float*  __restrict__ scores,
                      const int*    __restrict__ labels,
                      const float*  __restrict__ live_g,
                      const unsigned int* __restrict__ offmax,
                      float* __restrict__ out, int B)
{
    __shared__ float4 s_box[N_ANCH];                 // boxes + label*offset
    __shared__ __align__(16) float s_live[N_ANCH];   // async-staged live scores
    __shared__ float  s_redv[32];
    __shared__ int    s_redi[32];
    __shared__ int    s_besti;
    __shared__ int    s_keep[MAX_DET];
    __shared__ float  s_ksc[MAX_DET];

    const int b   = blockIdx.x;
    const int tid = threadIdx.x;
    const float offset = __uint_as_float(offmax[b]) + 1.0f;

    const float4* bx = boxes  + (size_t)b * N_ANCH;
    const int*    lb = labels + (size_t)b * N_ANCH;
    const float*  lv = live_g + (size_t)b * N_ANCH;

    // ---- stage raw boxes + live into LDS (async DMA on gfx1250) ----
#ifdef YOLOX_ASYNC_LDS
#pragma unroll
    for (int s = 0; s < CHUNK; ++s) {
        int j = tid + s * NMS_THREADS;
        if (j < N_ANCH)
            __builtin_amdgcn_global_load_async_to_lds_b128(
                yolox_to_global(bx + j), yolox_to_lds(&s_box[j]), 0, 0);
    }
    for (int j = tid * 4; j < N_ANCH; j += NMS_THREADS * 4)
        __builtin_amdgcn_global_load_async_to_lds_b128(
            yolox_to_global(lv + j), yolox_to_lds(&s_live[j]), 0, 0);
#if __has_builtin(__builtin_amdgcn_s_wait_asynccnt)
    __builtin_amdgcn_s_wait_asynccnt(0);
#else
    asm volatile("s_wait_asynccnt 0" ::: "memory");
#endif
#else
    for (int j = tid; j < N_ANCH; j += NMS_THREADS) { s_box[j] = bx[j]; s_live[j] = lv[j]; }
#endif
    __syncthreads();

    // ---- register-resident per-thread state: add class offset in place ----
    float  lv_r[CHUNK];
    float4 bx_r[CHUNK];
#pragma unroll
    for (int s = 0; s < CHUNK; ++s) {
        int j = tid + s * NMS_THREADS;
        if (j < N_ANCH) {
            float4 v = s_box[j];
            float  o = (float)lb[j] * offset;
            float4 vo = make_float4(v.x + o, v.y + o, v.z + o, v.w + o);
            bx_r[s] = vo;
            s_box[j] = vo;                 // LDS copy used for best-box lookup
            lv_r[s] = s_live[j];
        } else {
            bx_r[s] = make_float4(0.f, 0.f, 0.f, 0.f);  // zero-area: IoU == 0
            lv_r[s] = NEG_INF;
        }
    }
    __syncthreads();

    // ---- 100 greedy iterations, 2 barriers each ----
    for (int it = 0; it < MAX_DET; ++it) {
        // local argmax over registers (ties -> smallest index, like jnp.argmax)
        float v = NEG_INF;
        int   idx = 0;
#pragma unroll
        for (int s = 0; s < CHUNK; ++s) {
            int j = tid + s * NMS_THREADS;
            float x = lv_r[s];
            if (x > v || (x == v && j < idx)) { v = x; idx = j; }
        }
        // wave32 shuffle reduce
#pragma unroll
        for (int off = 16; off; off >>= 1) {
            float ov = __shfl_down(v, off);
            int   oi = __shfl_down(idx, off);
            if (ov > v || (ov == v && oi < idx)) { v = ov; idx = oi; }
        }
        const int wid = tid >> 5, lane = tid & 31;
        if (lane == 0) { s_redv[wid] = v; s_redi[wid] = idx; }
        __syncthreads();
        if (wid == 0) {
            v = s_redv[lane]; idx = s_redi[lane];
#pragma unroll
            for (int off = 16; off; off >>= 1) {
                float ov = __shfl_down(v, off);
                int   oi = __shfl_down(idx, off);
                if (ov > v || (ov == v && oi < idx)) { v = ov; idx = oi; }
            }
            if (lane == 0) { s_besti = idx; s_keep[it] = idx; s_ksc[it] = v; }
        }
        __syncthreads();

        // broadcast best box from LDS (same-address ds_load broadcasts)
        const int    best  = s_besti;
        const float4 bb    = s_box[best];
        const float  a1    = (bb.z - bb.x) * (bb.w - bb.y);
        const int    owner = best & (NMS_THREADS - 1);
        const int    slot  = best >> 10;   // log2(NMS_THREADS)

        // suppress in registers; owning thread also forces live[best] = -inf
#pragma unroll
        for (int s = 0; s < CHUNK; ++s) {
            float4 c  = bx_r[s];
            float iw  = fmaxf(fminf(bb.z, c.z) - fmaxf(bb.x, c.x), 0.0f);
            float ih  = fmaxf(fminf(bb.w, c.w) - fmaxf(bb.y, c.y), 0.0f);
            float inter = iw * ih;
            float a2  = (c.z - c.x) * (c.w - c.y);
            float iou = inter / (a1 + a2 - inter + 1e-6f);
            if ((iou > NMS_THR) || ((tid == owner) && (s == slot)))
                lv_r[s] = NEG_INF;
        }
        // next iteration's s_redv/s_besti writes are fenced by its own barriers
    }
    __syncthreads();

    // ---- emit: (boxes[B,100,4], scores[B,100], labels[B,100], valid[B,100])
    if (tid < MAX_DET) {
        const int   k     = s_keep[tid];
        const float ksc   = s_ksc[tid];
        const bool  valid = (ksc > NEG_INF);

        float4* out_boxes  = (float4*)out;                       // B*100 float4
        float*  out_scores = out + (size_t)B * MAX_DET * 4;
        float*  out_labels = out_scores + (size_t)B * MAX_DET;
        float*  out_valid  = out_labels + (size_t)B * MAX_DET;

        const int base = b * MAX_DET + tid;
        out_boxes[base]  = valid ? bx[k] : make_float4(0.f, 0.f, 0.f, 0.f);
        out_scores[base] = valid ? scores[(size_t)b * N_ANCH + k] : 0.0f;
        out_labels[base] = valid ? (float)lb[k] : -1.0f;
        out_valid[base]  = valid ? 1.0f : 0.0f;
    }
}

// ------------------------------- launcher -----------------------------------
extern "C" void kernel_launch(void* const* d_in, const int* in_sizes, int n_in,
                              void* d_out, int out_size, void* d_ws, size_t ws_size,
                              hipStream_t stream)
{
    // setup_inputs() dict order: cls0, reg0, obj0, cls1, reg1, obj1, cls2, reg2, obj2
    const float* cls0 = (const float*)d_in[0];
    const float* reg0 = (const float*)d_in[1];
    const float* obj0 = (const float*)d_in[2];
    const float* cls1 = (const float*)d_in[3];
    const float* reg1 = (const float*)d_in[4];
    const float* obj1 = (const float*)d_in[5];
    const float* cls2 = (const float*)d_in[6];
    const float* reg2 = (const float*)d_in[7];
    const float* obj2 = (const float*)d_in[8];

    const int B = in_sizes[2] / (80 * 80);   // obj0 is (B,1,80,80)

    // workspace layout (needs 256 + B*8400*28 bytes ~= 7.5MB @ B=32)
    unsigned int* offmax = (unsigned int*)d_ws;
    float4* boxes  = (float4*)((char*)d_ws + 256);
    float*  scores = (float*)(boxes + (size_t)B * N_ANCH);
    int*    labels = (int*)(scores + (size_t)B * N_ANCH);
    float*  live   = (float*)(labels + (size_t)B * N_ANCH);

    yolox_init_kernel<<<dim3(1), dim3(64), 0, stream>>>(offmax, B);

    yolox_decode_kernel<<<dim3(25, B), dim3(256), 0, stream>>>(
        cls0, reg0, obj0, 6400, 80, 8.0f, 0, boxes, scores, labels, live, offmax);
    yolox_decode_kernel<<<dim3(7, B), dim3(256), 0, stream>>>(
        cls1, reg1, obj1, 1600, 40, 16.0f, 6400, boxes, scores, labels, live, offmax);
    yolox_decode_kernel<<<dim3(2, B), dim3(256), 0, stream>>>(
        cls2, reg2, obj2, 400, 20, 32.0f, 8000, boxes, scores, labels, live, offmax);

    yolox_nms_kernel<<<dim3(B), dim3(NMS_THREADS), 0, stream>>>(
        boxes, scores, labels, live, offmax, (float*)d_out, B);

    (void)n_in; (void)out_size; (void)ws_size;
}